// Model_50379966382551
// MI455X (gfx1250) — compile-verified
//
#include <hip/hip_runtime.h>
#include <hip/hip_bf16.h>

// Segmented top-4 over CSR segments: one block = 256 consecutive segments.
// The block's contiguous edge window is staged into LDS with CDNA5 async
// global->LDS B128 copies (ASYNCcnt-tracked, double buffered); each thread
// keeps a register top-4 for its segment and scans from LDS.

#define BLK        256
#define CHUNK      4096                     // floats per staged chunk (16 KB)
#define VEC        4                        // floats per b128 async op
#define PER_THREAD (CHUNK / (BLK * VEC))    // 4 b128 issues per thread per chunk
#define TOPK       4

typedef int v4i __attribute__((vector_size(4 * sizeof(int))));

// ---------- CDNA5 async global->LDS copies (tracked by ASYNCcnt) ----------
__device__ __forceinline__ void async_g2l_b128(float* lds_dst, const float* g_src) {
#if __has_builtin(__builtin_amdgcn_global_load_async_to_lds_b128)
    __builtin_amdgcn_global_load_async_to_lds_b128(
        (__attribute__((address_space(1))) v4i*)((v4i*)g_src),
        (__attribute__((address_space(3))) v4i*)((v4i*)lds_dst),
        /*imm offset*/ 0, /*cpol*/ 0);
#else
    asm volatile("global_load_async_to_lds_b128 %0, %1, off"
                 :
                 : "v"((__attribute__((address_space(3))) v4i*)((v4i*)lds_dst)),
                   "v"((__attribute__((address_space(1))) v4i*)((v4i*)g_src))
                 : "memory");
#endif
}

#if __has_builtin(__builtin_amdgcn_s_wait_asynccnt)
#define WAIT_ASYNCCNT(n) __builtin_amdgcn_s_wait_asynccnt(n)
#else
#define WAIT_ASYNCCNT(n) asm volatile("s_wait_asynccnt %0" ::"i"(n) : "memory")
#endif

// Stable top-4 insertion: strict '>' keeps earlier (lower-index) entries ahead
// on ties, matching the reference's lexsort((-score, seg)) stability.
__device__ __forceinline__ void insert4(float s, int e,
                                        float& v0, float& v1, float& v2, float& v3,
                                        int& i0, int& i1, int& i2, int& i3) {
    if (s > v3) {
        if (s > v2) {
            if (s > v1) {
                if (s > v0) {
                    v3 = v2; i3 = i2; v2 = v1; i2 = i1; v1 = v0; i1 = i0;
                    v0 = s;  i0 = e;
                } else {
                    v3 = v2; i3 = i2; v2 = v1; i2 = i1;
                    v1 = s;  i1 = e;
                }
            } else {
                v3 = v2; i3 = i2;
                v2 = s;  i2 = e;
            }
        } else {
            v3 = s; i3 = e;
        }
    }
}

__global__ __launch_bounds__(BLK) void seg_topk4_kernel(
    const int* __restrict__ row_ptr,
    const float* __restrict__ scores,
    float* __restrict__ out_vals,
    int* __restrict__ out_idx,
    int nseg, int nedge)
{
    __shared__ __align__(16) float buf[2][CHUNK];

    const int tid    = threadIdx.x;
    const int seg0   = blockIdx.x * BLK;
    const int my_seg = seg0 + tid;
    const bool active = (my_seg < nseg);

    // Block-contiguous edge window (CSR => contiguous in memory).
    const int blk_begin = row_ptr[seg0];
    const int last_seg  = (seg0 + BLK < nseg) ? (seg0 + BLK) : nseg;
    const int blk_end   = row_ptr[last_seg];

    int my_s = 0, my_e = 0;
    if (active) { my_s = row_ptr[my_seg]; my_e = row_ptr[my_seg + 1]; }

    const float NEG_INF = -__builtin_inff();
    float v0 = NEG_INF, v1 = NEG_INF, v2 = NEG_INF, v3 = NEG_INF;
    int   i0 = -1, i1 = -1, i2 = -1, i3 = -1;

    // Stage from a 16-byte-aligned base; leading garbage elements sit below
    // every thread's [my_s, my_e) window and are never read.
    const int abase = blk_begin & ~(VEC - 1);
    const int gmax  = nedge - VEC;          // E is a multiple of 4 => aligned

    // Issue one full chunk: uniform 4 b128 copies per thread so an immediate
    // s_wait_asynccnt 4 fences the previously issued chunk.
    auto issue = [&](int c) {
        const int cbase = abase + c * CHUNK;
        float* dst = buf[c & 1];
#pragma unroll
        for (int j = 0; j < PER_THREAD; ++j) {
            const int k = (j * BLK + tid) * VEC;   // coalesced, 16B-aligned
            int g = cbase + k;
            if (g > gmax) g = gmax;                // tail clamp: dup loads, harmless
            async_g2l_b128(&dst[k], &scores[g]);
        }
    };

    const int total   = blk_end - abase;
    const int nchunks = (total + CHUNK - 1) / CHUNK;

    if (nchunks > 0) issue(0);

    for (int c = 0; c < nchunks; ++c) {
        const bool more = (c + 1) < nchunks;
        if (more) issue(c + 1);                // prefetch next chunk (double buffer)
        if (more) { WAIT_ASYNCCNT(PER_THREAD); } else { WAIT_ASYNCCNT(0); }
        __syncthreads();                       // all waves' staging complete

        const int cbase = abase + c * CHUNK;
        const int cend  = (cbase + CHUNK < blk_end) ? (cbase + CHUNK) : blk_end;
        const int lo = (my_s > cbase) ? my_s : cbase;
        const int hi = (my_e < cend) ? my_e : cend;

        const float* b = buf[c & 1];
        for (int e = lo; e < hi; ++e) {
            insert4(b[e - cbase], e, v0, v1, v2, v3, i0, i1, i2, i3);
        }
        __syncthreads();                       // done reading before re-stage
    }

    if (active) {
        float* ov = out_vals + (size_t)my_seg * TOPK;
        ov[0] = v0; ov[1] = v1; ov[2] = v2; ov[3] = v3;
        int* oi = out_idx + (size_t)my_seg * TOPK;
        oi[0] = i0; oi[1] = i1; oi[2] = i2; oi[3] = i3;
    }
}

extern "C" void kernel_launch(void* const* d_in, const int* in_sizes, int n_in,
                              void* d_out, int out_size, void* d_ws, size_t ws_size,
                              hipStream_t stream) {
    const int*   row_ptr = (const int*)d_in[0];
    const float* scores  = (const float*)d_in[1];
    const int nseg  = in_sizes[0] - 1;   // row_ptr has N+1 entries
    const int nedge = in_sizes[1];

    // Tuple output: [topk_vals (N*4 f32)] then [topk_idx (N*4 i32)] flat.
    float* out_vals = (float*)d_out;
    int*   out_idx  = (int*)d_out + (size_t)nseg * TOPK;

    const int grid = (nseg + BLK - 1) / BLK;
    seg_topk4_kernel<<<grid, BLK, 0, stream>>>(row_ptr, scores, out_vals, out_idx,
                                               nseg, nedge);
}